// BANLayer_90606630076421
// MI455X (gfx1250) — compile-verified
//
#include <hip/hip_runtime.h>
#include <hip/hip_bf16.h>
#include <math.h>

typedef __attribute__((ext_vector_type(2))) float v2f;
typedef __attribute__((ext_vector_type(8))) float v8f;

#define B_SZ   8
#define NV     128
#define NQ     512
#define HD     256   // H_DIM
#define HO     2     // H_OUT
#define HN     512   // H_OUT*H_DIM

// ---------------------------------------------------------------------------
// CDNA5 async global->LDS copy (GLOBAL_LOAD_ASYNC_TO_LDS_B128, ASYNCcnt).
// lds_off: wave-relative LDS byte offset (low 32 bits of generic pointer).
// ---------------------------------------------------------------------------
__device__ __forceinline__ void async_copy_b128(unsigned lds_off, const float* gptr) {
    asm volatile("global_load_async_to_lds_b128 %0, %1, off"
                 :: "v"(lds_off), "v"(gptr) : "memory");
}
__device__ __forceinline__ void wait_async0() {
    asm volatile("s_wait_asynccnt 0" ::: "memory");
}

// ---------------------------------------------------------------------------
// Projection: Y[M x 512] = X[M x 128] @ W[128 x 512] + bias[512]
// All 8 waves of a block share one A tile (tm uniform): stage it to LDS via
// async-to-LDS once, then each wave runs its own 16x16 WMMA tile over K=128.
// ---------------------------------------------------------------------------
__global__ void proj_kernel(const float* __restrict__ X,
                            const float* __restrict__ W,
                            const float* __restrict__ bias,
                            float* __restrict__ Y) {
    __shared__ __align__(16) float sA[16 * 128];   // 8 KB A tile

    const int tid  = threadIdx.x;
    const int wave = tid >> 5;
    const int lane = tid & 31;
    const int idx  = lane & 15;       // M index (A) / N index (B,C,D)
    const int hl   = lane >> 4;       // lane half
    const int kb   = hl * 2;          // K sub-offset within frag

    const int t  = blockIdx.x * 8 + wave;
    const int tm = t >> 5;            // block-uniform (8 | 32 N-tiles)
    const int tn = t & 31;

    // Stage A tile: 16 rows x 128 K = 2048 floats; 256 thr x 2 x b128.
    {
        const float* g = X + (size_t)tm * 16 * 128 + tid * 4;
        unsigned l = (unsigned)(uintptr_t)&sA[tid * 4];
        async_copy_b128(l, g);
        async_copy_b128(l + 4096, g + 1024);
    }
    wait_async0();
    __syncthreads();

    const float* aRow = &sA[idx * 128];
    const float* bCol = W + tn * 16 + idx;

    v8f acc = {};
#pragma unroll 8
    for (int k0 = 0; k0 < 128; k0 += 4) {
        v2f a = *(const v2f*)(aRow + k0 + kb);          // ds_load_b64
        v2f b;
        b.x = bCol[(size_t)(k0 + kb) * 512];
        b.y = bCol[(size_t)(k0 + kb + 1) * 512];
        acc = __builtin_amdgcn_wmma_f32_16x16x4_f32(false, a, false, b,
                                                    (short)0, acc, false, false);
    }

    const float bb = bias[tn * 16 + idx];
#pragma unroll
    for (int r = 0; r < 8; ++r) {
        Y[(size_t)(tm * 16 + r + 8 * hl) * 512 + tn * 16 + idx] = acc[r] + bb;
    }
}

// ---------------------------------------------------------------------------
// score[b,h,i,j] = sum_d v_[b,i,h,d]*wa[d]*q_[b,j,h,d] + ba, then softmax
// over i (the 128-row axis) per column j.  One block per (b, h, 64-col tile).
// ---------------------------------------------------------------------------
__global__ void score_softmax_kernel(const float* __restrict__ v_,
                                     const float* __restrict__ q_,
                                     const float* __restrict__ wa,
                                     const float* __restrict__ ba_ptr,
                                     float* __restrict__ att) {
    __shared__ float s_score[128][64];

    const int bh_jt = blockIdx.x;          // [0, 128)
    const int b  = bh_jt >> 4;
    const int h  = (bh_jt >> 3) & 1;
    const int jt = bh_jt & 7;              // 64-column tile of j
    const int bh = b * HO + h;

    const int wave = threadIdx.x >> 5;
    const int lane = threadIdx.x & 31;
    const int idx  = lane & 15;
    const int hl   = lane >> 4;
    const int kb   = hl * 2;

    const float bav = *ba_ptr;

    // 8 M-tiles x 4 N-tiles = 32 tiles; 4 per wave (uniform loop -> EXEC all 1s)
    for (int wt = wave; wt < 32; wt += 8) {
        const int tm = wt >> 2;
        const int tn = wt & 3;

        const float* aBase = v_ + ((size_t)(b * NV + tm * 16 + idx) * HO + h) * HD;
        const float* bBase = q_ + ((size_t)(b * NQ + jt * 64 + tn * 16 + idx) * HO + h) * HD;

        v8f acc = {};
#pragma unroll 8
        for (int k0 = 0; k0 < HD; k0 += 4) {
            v2f a = *(const v2f*)(aBase + k0 + kb);
            v2f w = *(const v2f*)(wa + k0 + kb);
            a *= w;                               // fuse the wa scale into A
            v2f bf = *(const v2f*)(bBase + k0 + kb);  // Qh^T contiguous in d
            acc = __builtin_amdgcn_wmma_f32_16x16x4_f32(false, a, false, bf,
                                                        (short)0, acc, false, false);
        }
#pragma unroll
        for (int r = 0; r < 8; ++r)
            s_score[tm * 16 + r + 8 * hl][tn * 16 + idx] = acc[r] + bav;
    }

    __syncthreads();

    // Column softmax over i = 0..127 (matches softmax(axis=2) in reference).
    const int c = threadIdx.x;
    if (c < 64) {
        float m = -INFINITY;
#pragma unroll 4
        for (int i = 0; i < 128; ++i) m = fmaxf(m, s_score[i][c]);
        float sum = 0.f;
#pragma unroll 4
        for (int i = 0; i < 128; ++i) {
            float e = __expf(s_score[i][c] - m);
            s_score[i][c] = e;
            sum += e;
        }
        const float inv = 1.0f / sum;
        float* outCol = att + (size_t)bh * NV * NQ + (size_t)jt * 64 + c;
#pragma unroll 4
        for (int i = 0; i < 128; ++i)
            outCol[(size_t)i * NQ] = s_score[i][c] * inv;
    }
}

// ---------------------------------------------------------------------------
// ctx[bh, i, d] = sum_j att[bh,i,j] * q_[b,j,h,d]
// Per (b,h): (128x512)@(512x256).  All 8 waves share one 16x512 att tile
// (tm block-uniform): async-stage it into 32 KB LDS once.
// ---------------------------------------------------------------------------
__global__ void ctx_kernel(const float* __restrict__ att,
                           const float* __restrict__ q_,
                           float* __restrict__ ctx) {
    __shared__ __align__(16) float sA[16 * 512];   // 32 KB att tile

    const int tid  = threadIdx.x;
    const int wave = tid >> 5;
    const int lane = tid & 31;
    const int idx  = lane & 15;
    const int hl   = lane >> 4;
    const int kb   = hl * 2;

    const int t   = blockIdx.x * 8 + wave;   // [0, 2048)
    const int bh  = t >> 7;                  // block-uniform
    const int rem = t & 127;
    const int tm  = rem >> 4;                // block-uniform (8 | 16 N-tiles)
    const int tn  = rem & 15;
    const int b   = bh >> 1;
    const int h   = bh & 1;

    // Stage att tile: 16 rows x 512 = 8192 floats; 256 thr x 8 x b128.
    {
        const float* g = att + ((size_t)bh * NV + tm * 16) * NQ + tid * 4;
        unsigned l = (unsigned)(uintptr_t)&sA[tid * 4];
#pragma unroll
        for (int it = 0; it < 8; ++it)
            async_copy_b128(l + it * 4096, g + it * 1024);
    }
    wait_async0();
    __syncthreads();

    const float* aRow  = &sA[idx * 512];
    const float* bBase = q_ + (size_t)b * NQ * HN + (size_t)h * HD + tn * 16 + idx;

    v8f acc = {};
#pragma unroll 4
    for (int k0 = 0; k0 < NQ; k0 += 4) {
        v2f a = *(const v2f*)(aRow + k0 + kb);          // ds_load_b64
        v2f bf;
        bf.x = bBase[(size_t)(k0 + kb) * HN];
        bf.y = bBase[(size_t)(k0 + kb + 1) * HN];
        acc = __builtin_amdgcn_wmma_f32_16x16x4_f32(false, a, false, bf,
                                                    (short)0, acc, false, false);
    }
#pragma unroll
    for (int r = 0; r < 8; ++r)
        ctx[((size_t)bh * NV + tm * 16 + r + 8 * hl) * HD + tn * 16 + idx] = acc[r];
}

// ---------------------------------------------------------------------------
// head[bh, d] = sum_i ctx[bh,i,d] * v_[b,i,h,d]      (4096 outputs)
// ---------------------------------------------------------------------------
__global__ void head_kernel(const float* __restrict__ ctx,
                            const float* __restrict__ v_,
                            float* __restrict__ head) {
    const int tid = blockIdx.x * 256 + threadIdx.x;  // [0, 4096)
    const int d  = tid & 255;
    const int bh = tid >> 8;
    const int b  = bh >> 1;
    const int h  = bh & 1;
    float s = 0.f;
#pragma unroll 4
    for (int i = 0; i < NV; ++i)
        s += ctx[((size_t)bh * NV + i) * HD + d] *
             v_[((size_t)(b * NV + i) * HO + h) * HD + d];
    head[tid] = s;
}

// ---------------------------------------------------------------------------
// fused[b, o] = sum_k head[b*512+k] * Wo[k, o] + bo[o]   (2048 outputs)
// ---------------------------------------------------------------------------
__global__ void fused_kernel(const float* __restrict__ head,
                             const float* __restrict__ Wo,
                             const float* __restrict__ bo,
                             float* __restrict__ out) {
    const int tid = blockIdx.x * 256 + threadIdx.x;  // [0, 2048)
    const int o = tid & 255;
    const int b = tid >> 8;
    float s = bo[o];
#pragma unroll 4
    for (int k = 0; k < HN; ++k)
        s += head[b * HN + k] * Wo[(size_t)k * HD + o];
    out[tid] = s;
}

// ---------------------------------------------------------------------------
extern "C" void kernel_launch(void* const* d_in, const int* in_sizes, int n_in,
                              void* d_out, int out_size, void* d_ws, size_t ws_size,
                              hipStream_t stream) {
    const float* v  = (const float*)d_in[0];
    const float* q  = (const float*)d_in[1];
    const float* Wv = (const float*)d_in[2];
    const float* bv = (const float*)d_in[3];
    const float* Wq = (const float*)d_in[4];
    const float* bq = (const float*)d_in[5];
    const float* wa = (const float*)d_in[6];
    const float* ba = (const float*)d_in[7];
    const float* Wo = (const float*)d_in[8];
    const float* bo = (const float*)d_in[9];

    float* out = (float*)d_out;
    float* att = out + (size_t)B_SZ * HD;          // fused occupies first 2048

    float* ws     = (float*)d_ws;
    float* ws_v   = ws;                            // 524288  f32 : v_ (B,NV,HO,HD)
    float* ws_q   = ws + 524288;                   // 2097152 f32 : q_ (B,NQ,HO,HD)
    float* ws_ctx = ws + 2621440;                  // 524288  f32 : ctx (B*HO,NV,HD)
    float* ws_hd  = ws + 3145728;                  // 4096    f32 : head

    // v_ projection: M=1024 -> 64*32 tiles -> 256 blocks of 8 waves
    proj_kernel<<<256, 256, 0, stream>>>(v, Wv, bv, ws_v);
    // q_ projection: M=4096 -> 256*32 tiles -> 1024 blocks
    proj_kernel<<<1024, 256, 0, stream>>>(q, Wq, bq, ws_q);
    // score + softmax -> att (written straight into d_out)
    score_softmax_kernel<<<128, 256, 0, stream>>>(ws_v, ws_q, wa, ba, att);
    // ctx = att @ Qh : 2048 tiles -> 256 blocks
    ctx_kernel<<<256, 256, 0, stream>>>(att, ws_q, ws_ctx);
    // head reduction
    head_kernel<<<16, 256, 0, stream>>>(ws_ctx, ws_v, ws_hd);
    // final fused output
    fused_kernel<<<8, 256, 0, stream>>>(ws_hd, Wo, bo, out);
}